// HashEncoder_82978768158951
// MI455X (gfx1250) — compile-verified
//
#include <hip/hip_runtime.h>

#define THREADS 256
#define RS 34          // LDS staging row stride in floats (34 -> conflict-free b64 writes)

#if defined(__has_builtin)
#  if __has_builtin(__builtin_amdgcn_global_load_async_to_lds_b32) && \
      __has_builtin(__builtin_amdgcn_s_wait_asynccnt)
#    define HAS_ASYNC_LDS 1
#  endif
#endif

__global__ __launch_bounds__(THREADS) void hashenc_kernel(
    const float* __restrict__ pos, const float2* __restrict__ tab,
    float* __restrict__ out, int n)
{
    constexpr unsigned P1 = 2654435761u, P2 = 805459861u;
    // level sizes: dense levels 0-4, hashed (2^19) levels 5-15; offsets = prefix sums
    constexpr unsigned kSize[16] = {
        4096u, 12168u, 29792u, 79512u, 205384u,
        524288u, 524288u, 524288u, 524288u, 524288u, 524288u,
        524288u, 524288u, 524288u, 524288u, 524288u };
    constexpr unsigned kOff[16] = {
        0u, 4096u, 16264u, 46056u, 125568u, 330952u,
        855240u, 1379528u, 1903816u, 2428104u, 2952392u,
        3476680u, 4000968u, 4525256u, 5049544u, 5573832u };
    // grid_scale = 16*2^(7l/15) - 1 (precomputed)
    constexpr float kScale[16] = {
        15.0f, 21.110606f, 29.554933f, 41.224247f, 57.350233f, 79.634947f,
        110.430472f, 152.987242f, 211.796997f, 293.066788f, 405.374668f,
        560.574452f, 775.046899f, 1071.429100f, 1481.008200f, 2047.0f };

    __shared__ float stage[THREADS * RS];   // [256 rows][32 feats + pad]
    __shared__ float spos[THREADS * 3];

    const int tid = threadIdx.x;
    const int blockBase = blockIdx.x * THREADS;

    // ---- stage this block's positions into LDS (gfx1250 async path if available) ----
    {
        const unsigned cap3  = (unsigned)n * 3u - 1u;
        const unsigned base3 = (unsigned)blockBase * 3u;
#if HAS_ASYNC_LDS
        typedef __attribute__((address_space(3))) int lint;
        typedef __attribute__((address_space(1))) int gint;
        lint* sdst = (lint*)spos;
        #pragma unroll
        for (int k = 0; k < 3; ++k) {
            unsigned idx = base3 + (unsigned)tid + (unsigned)(k * THREADS);
            if (idx > cap3) idx = cap3;
            __builtin_amdgcn_global_load_async_to_lds_b32(
                (gint*)(pos + idx), sdst + tid + k * THREADS, 0, 0);
        }
        __builtin_amdgcn_s_wait_asynccnt(0);
#else
        #pragma unroll
        for (int k = 0; k < 3; ++k) {
            unsigned idx = base3 + (unsigned)tid + (unsigned)(k * THREADS);
            if (idx > cap3) idx = cap3;
            spos[tid + k * THREADS] = pos[idx];
        }
#endif
        __syncthreads();
    }

    int lt = tid;
    if (blockBase + lt >= n) lt = (n - 1) - blockBase;  // clamp for partial tail block
    const float x = spos[lt * 3 + 0];
    const float y = spos[lt * 3 + 1];
    const float z = spos[lt * 3 + 2];

    // ---- 16 levels, fully unrolled: incremental hash + 8-corner gather-blend ----
    #pragma unroll
    for (int l = 0; l < 16; ++l) {
        const float    s   = kScale[l];
        const unsigned sz  = kSize[l];    // compile-time: pow2 -> AND, else magic div
        const unsigned off = kOff[l];

        const float px = fmaf(x, s, 0.5f);
        const float py = fmaf(y, s, 0.5f);
        const float pz = fmaf(z, s, 0.5f);
        const float gx = floorf(px), gy = floorf(py), gz = floorf(pz);
        const float ax = px - gx, ay = py - gy, az = pz - gz;   // trilinear fracs
        const float bx = 1.0f - ax, by = 1.0f - ay, bz = 1.0f - az;
        const unsigned ix = (unsigned)(int)gx;
        const unsigned iy = (unsigned)(int)gy;
        const unsigned iz = (unsigned)(int)gz;

        const unsigned hx0 = ix,       hx1 = ix + 1u;
        const unsigned hy0 = iy * P1,  hy1 = hy0 + P1;
        const unsigned hz0 = iz * P2,  hz1 = hz0 + P2;

        // 8 hashed indices (corner bit layout: x=bit2, y=bit1, z=bit0)
        const unsigned h0 = ((hx0 ^ hy0 ^ hz0) % sz) + off;  // 000
        const unsigned h1 = ((hx0 ^ hy0 ^ hz1) % sz) + off;  // 001
        const unsigned h2 = ((hx0 ^ hy1 ^ hz0) % sz) + off;  // 010
        const unsigned h3 = ((hx0 ^ hy1 ^ hz1) % sz) + off;  // 011
        const unsigned h4 = ((hx1 ^ hy0 ^ hz0) % sz) + off;  // 100
        const unsigned h5 = ((hx1 ^ hy0 ^ hz1) % sz) + off;  // 101
        const unsigned h6 = ((hx1 ^ hy1 ^ hz0) % sz) + off;  // 110
        const unsigned h7 = ((hx1 ^ hy1 ^ hz1) % sz) + off;  // 111

        // issue all 8 gathers before consuming (MLP; L2-resident table)
        const float2 t0 = tab[h0];
        const float2 t1 = tab[h1];
        const float2 t2 = tab[h2];
        const float2 t3 = tab[h3];
        const float2 t4 = tab[h4];
        const float2 t5 = tab[h5];
        const float2 t6 = tab[h6];
        const float2 t7 = tab[h7];

        float f0 = 0.0f, f1 = 0.0f, w;
        w = bx * (by * bz); f0 = fmaf(w, t0.x, f0); f1 = fmaf(w, t0.y, f1);
        w = bx * (by * az); f0 = fmaf(w, t1.x, f0); f1 = fmaf(w, t1.y, f1);
        w = bx * (ay * bz); f0 = fmaf(w, t2.x, f0); f1 = fmaf(w, t2.y, f1);
        w = bx * (ay * az); f0 = fmaf(w, t3.x, f0); f1 = fmaf(w, t3.y, f1);
        w = ax * (by * bz); f0 = fmaf(w, t4.x, f0); f1 = fmaf(w, t4.y, f1);
        w = ax * (by * az); f0 = fmaf(w, t5.x, f0); f1 = fmaf(w, t5.y, f1);
        w = ax * (ay * bz); f0 = fmaf(w, t6.x, f0); f1 = fmaf(w, t6.y, f1);
        w = ax * (ay * az); f0 = fmaf(w, t7.x, f0); f1 = fmaf(w, t7.y, f1);

        // stage per-level result: ds_store_b64, conflict-free (stride 34 floats)
        *(float2*)&stage[tid * RS + 2 * l] = make_float2(f0, f1);
    }

    __syncthreads();

    // ---- coalesced non-temporal write-out: keep the hash table resident in L2 ----
    const size_t obase = (size_t)blockBase * 32;
    const size_t total = (size_t)n * 32;
    #pragma unroll
    for (int i = 0; i < 32; ++i) {
        const int e = i * THREADS + tid;      // wave reads one full LDS row, writes 128B line
        const size_t o = obase + (size_t)e;
        if (o < total) {
            const float v = stage[(e >> 5) * RS + (e & 31)];
            __builtin_nontemporal_store(v, out + o);
        }
    }
}

extern "C" void kernel_launch(void* const* d_in, const int* in_sizes, int n_in,
                              void* d_out, int out_size, void* d_ws, size_t ws_size,
                              hipStream_t stream) {
    const float*  pos = (const float*)d_in[0];
    const float2* tab = (const float2*)d_in[1];
    float* out = (float*)d_out;
    const int n = in_sizes[0] / 3;                 // 2,097,152 positions
    const int blocks = (n + THREADS - 1) / THREADS;
    hashenc_kernel<<<blocks, THREADS, 0, stream>>>(pos, tab, out, n);
}